// DistanceTransform_14800457302627
// MI455X (gfx1250) — compile-verified
//
#include <hip/hip_runtime.h>

// EDT skeleton/boundary for MI455X (gfx1250, wave32).
// d_in[0]: masks fp32 [16,512,512].  d_out: skeleton fp32 [16,512,512] ++ boundary fp32 [16,512,512].
// d_ws: 64 bytes used (16 per-batch max slots).

#define Bn 16
#define Hn 512
#define Wn 512
#define HALF (Bn * Hn * Wn)   // 4194304 floats per output array

typedef __attribute__((ext_vector_type(4))) unsigned int u32x4;
typedef __attribute__((ext_vector_type(8))) int          i32x8;
typedef __attribute__((ext_vector_type(4))) int          i32x4;

// Probe: require the TDM builtin in the DEVICE pass only (host pass never has it).
#if defined(__HIP_DEVICE_COMPILE__)
#if !__has_builtin(__builtin_amdgcn_tensor_load_to_lds)
#error "gfx1250 TDM builtin __builtin_amdgcn_tensor_load_to_lds not available on this toolchain"
#endif
#define USE_TDM 1
#else
#define USE_TDM 0
#endif

// ---------------------------------------------------------------- init max
__global__ void k_init(unsigned int* mx) {
  if (threadIdx.x < Bn) mx[threadIdx.x] = 0u;
}

// ---------------------------------------------------------------- pass 1: rows
// One wave32 per row of 512; lane owns 16 contiguous elements.
// fwd[w] = w - prefixmax(p), p[j] = j if m[j]==0 else -(INF+1)   (INF = H+W = 1024)
// bwd[w] = suffixmin(q) - w, q[j] = j if m[j]==0 else  W+INF
// f2 = min(fwd,bwd)^2  -> stored to scratch (boundary half of d_out).
__global__ __launch_bounds__(256) void k_rowpass(const float* __restrict__ masks,
                                                 float* __restrict__ f2) {
  const int row  = blockIdx.x * 8 + (threadIdx.x >> 5);   // 1024 blocks * 8 waves = 8192 rows
  const int lane = threadIdx.x & 31;
  const float NEGI = -1025.0f;   // -(INF+1)
  const float POSI = 1536.0f;    //  W+INF

  const float4* rp4 = (const float4*)(masks + (size_t)row * Wn + lane * 16);
  float m[16];
#pragma unroll
  for (int t = 0; t < 4; ++t) {
    float4 v = rp4[t];
    m[4*t+0] = v.x; m[4*t+1] = v.y; m[4*t+2] = v.z; m[4*t+3] = v.w;
  }
  const float bw = (float)(lane * 16);

  // in-lane prefix max of p
  float fa[16], pm = NEGI;
#pragma unroll
  for (int k = 0; k < 16; ++k) { float w = bw + k; pm = (m[k] == 0.0f) ? w : pm; fa[k] = pm; }
  // wave32 inclusive scan (max), then exclusive
  float run = pm;
#pragma unroll
  for (int off = 1; off < 32; off <<= 1) {
    float v = __shfl_up(run, off, 32);
    if (lane >= off) run = fmaxf(run, v);
  }
  float exl = __shfl_up(run, 1, 32);
  if (lane == 0) exl = NEGI;

  // in-lane suffix min of q
  float ba[16], sm = POSI;
#pragma unroll
  for (int k = 15; k >= 0; --k) { float w = bw + k; sm = (m[k] == 0.0f) ? w : sm; ba[k] = sm; }
  float run2 = sm;
#pragma unroll
  for (int off = 1; off < 32; off <<= 1) {
    float v = __shfl_down(run2, off, 32);
    if (lane + off < 32) run2 = fminf(run2, v);
  }
  float exr = __shfl_down(run2, 1, 32);
  if (lane == 31) exr = POSI;

  float o[16];
#pragma unroll
  for (int k = 0; k < 16; ++k) {
    float w   = bw + k;
    float fwd = w - fmaxf(fa[k], exl);
    float bwd = fminf(ba[k], exr) - w;
    float d   = fminf(fwd, bwd);
    o[k] = d * d;                       // exact: integers < 2^24
  }
  float4* op4 = (float4*)(f2 + (size_t)row * Wn + lane * 16);
#pragma unroll
  for (int t = 0; t < 4; ++t) {
    float4 v; v.x = o[4*t]; v.y = o[4*t+1]; v.z = o[4*t+2]; v.w = o[4*t+3];
    op4[t] = v;
  }
}

// ---------------------------------------------------------------- pass 2: columns
// Tropical GEMM: edt2[i,w] = min_j f2[j,w] + (i-j)^2.
// Block = (b, 128-row i-tile, 64-col w-tile), 512 threads.
// Thread -> 2 rows (i, i+64) x 8 columns = 16 accumulators fed by the same LDS reads.
// The 512x64 f2 slab (128 KB) is DMA'd to LDS by the Tensor Data Mover.
// j unrolled x2 so acc updates fuse to v_min3_num_f32 (2 add + 1 min3 per 2 updates).
__global__ __launch_bounds__(512) void k_colpass(const float* __restrict__ f2,
                                                 float* __restrict__ edt2,
                                                 unsigned int* __restrict__ mx) {
  extern __shared__ float tile[];                       // 512*64 floats = 128 KB
  const int b  = blockIdx.z;
  const int i0 = blockIdx.y * 128;
  const int w0 = blockIdx.x * 64;

  if (threadIdx.x < 32) {                               // wave 0 issues the TDM
#if USE_TDM
    const unsigned long long ga =
        (unsigned long long)(f2 + ((size_t)b * Hn) * Wn + w0);   // tile start (row 0, col w0)
    const unsigned int ldsa = (unsigned int)(unsigned long long)(void*)tile; // LDS byte offset

    u32x4 g0 = { 1u,                                    // count=1, user descriptor
                 ldsa,                                  // lds_addr [63:32]
                 (unsigned int)ga,                      // global_addr low
                 ((unsigned int)(ga >> 32) & 0x01FFFFFFu) | (2u << 30) }; // addr hi | type=2
    i32x8 g1 = { (int)0x00020000,                       // data_size=2 -> 4-byte elements
                 (int)(512u << 16),                     // tensor_dim0 = 512 (lo16 in [63:48])
                 (int)(512u << 16),                     // tensor_dim0 hi=0 | tensor_dim1 = 512
                 (int)(64u  << 16),                     // tensor_dim1 hi=0 | tile_dim0 = 64
                 (int)512,                              // tile_dim1 = 512, tile_dim2 = 0
                 (int)512,                              // tensor_dim0_stride = 512 (lo32)
                 0, 0 };                                // stride hi, tensor_dim1_stride = 0
    i32x4 gz = { 0, 0, 0, 0 };
#if defined(__clang_major__) && (__clang_major__ >= 23)
    i32x8 gz8 = { 0, 0, 0, 0, 0, 0, 0, 0 };
    __builtin_amdgcn_tensor_load_to_lds(g0, g1, gz, gz, gz8, 0);
#else
    __builtin_amdgcn_tensor_load_to_lds(g0, g1, gz, gz, 0);
#endif
    __builtin_amdgcn_s_wait_tensorcnt(0);
#else
    // host-pass placeholder only (never executed; device pass requires the builtin)
    for (int t = (int)threadIdx.x; t < Hn * 64; t += 32)
      tile[t] = f2[((size_t)b * Hn + (t >> 6)) * Wn + w0 + (t & 63)];
#endif
  }
  __syncthreads();

  const int wg = threadIdx.x & 7;        // 8-column group
  const int iA = i0 + (threadIdx.x >> 3);
  const int iB = iA + 64;

  float4 a0 = { 1e30f, 1e30f, 1e30f, 1e30f };
  float4 a1 = a0, c0 = a0, c1 = a0;      // a*: row iA, c*: row iB
  float  sA = (float)iA;                 // sA = iA - j
  float  sB = (float)iB;                 // sB = iB - j
  const float4* tp = (const float4*)(tile + wg * 8);
#pragma unroll 2
  for (int j = 0; j < Hn; j += 2) {
    const float4* r = tp + j * 16;
    float4 t0 = r[0],  t1 = r[1];        // row j
    float4 u0 = r[16], u1 = r[17];       // row j+1
    float pA0 = sA * sA;  float qA = sA - 1.0f;  float pA1 = qA * qA;
    float pB0 = sB * sB;  float qB = sB - 1.0f;  float pB1 = qB * qB;

    a0.x = fminf(fminf(a0.x, pA0 + t0.x), pA1 + u0.x);
    a0.y = fminf(fminf(a0.y, pA0 + t0.y), pA1 + u0.y);
    a0.z = fminf(fminf(a0.z, pA0 + t0.z), pA1 + u0.z);
    a0.w = fminf(fminf(a0.w, pA0 + t0.w), pA1 + u0.w);
    a1.x = fminf(fminf(a1.x, pA0 + t1.x), pA1 + u1.x);
    a1.y = fminf(fminf(a1.y, pA0 + t1.y), pA1 + u1.y);
    a1.z = fminf(fminf(a1.z, pA0 + t1.z), pA1 + u1.z);
    a1.w = fminf(fminf(a1.w, pA0 + t1.w), pA1 + u1.w);

    c0.x = fminf(fminf(c0.x, pB0 + t0.x), pB1 + u0.x);
    c0.y = fminf(fminf(c0.y, pB0 + t0.y), pB1 + u0.y);
    c0.z = fminf(fminf(c0.z, pB0 + t0.z), pB1 + u0.z);
    c0.w = fminf(fminf(c0.w, pB0 + t0.w), pB1 + u0.w);
    c1.x = fminf(fminf(c1.x, pB0 + t1.x), pB1 + u1.x);
    c1.y = fminf(fminf(c1.y, pB0 + t1.y), pB1 + u1.y);
    c1.z = fminf(fminf(c1.z, pB0 + t1.z), pB1 + u1.z);
    c1.w = fminf(fminf(c1.w, pB0 + t1.w), pB1 + u1.w);

    sA -= 2.0f;
    sB -= 2.0f;
  }
  const size_t oA = ((size_t)b * Hn + iA) * Wn + w0 + wg * 8;
  const size_t oB = ((size_t)b * Hn + iB) * Wn + w0 + wg * 8;
  *(float4*)(edt2 + oA)     = a0;
  *(float4*)(edt2 + oA + 4) = a1;
  *(float4*)(edt2 + oB)     = c0;
  *(float4*)(edt2 + oB + 4) = c1;

  // per-batch max of edt2 (non-negative: uint compare == float compare)
  float tm = fmaxf(fmaxf(fmaxf(a0.x, a0.y), fmaxf(a0.z, a0.w)),
                   fmaxf(fmaxf(a1.x, a1.y), fmaxf(a1.z, a1.w)));
  tm = fmaxf(tm, fmaxf(fmaxf(fmaxf(c0.x, c0.y), fmaxf(c0.z, c0.w)),
                       fmaxf(fmaxf(c1.x, c1.y), fmaxf(c1.z, c1.w))));
#pragma unroll
  for (int off = 16; off; off >>= 1) tm = fmaxf(tm, __shfl_xor(tm, off, 32));
  if ((threadIdx.x & 31) == 0) atomicMax(mx + b, __float_as_uint(tm));
}

// ---------------------------------------------------------------- pass 3: normalize
__global__ __launch_bounds__(256) void k_norm(const float* __restrict__ masks,
                                              const unsigned int* __restrict__ mx,
                                              float* __restrict__ skel,   // in: edt2, out: skeleton
                                              float* __restrict__ bnd) {  // out: boundary
  const int idx = blockIdx.x * 256 + threadIdx.x;
  const int b   = idx >> 18;                       // / (512*512)
  const float mxd = sqrtf(__uint_as_float(mx[b])); // max dt = sqrt(max edt2)
  const float sk  = sqrtf(skel[idx]) / mxd;        // matches reference: dt / mx
  skel[idx] = sk;
  bnd[idx]  = masks[idx] - sk;
}

// ---------------------------------------------------------------- launcher
extern "C" void kernel_launch(void* const* d_in, const int* in_sizes, int n_in,
                              void* d_out, int out_size, void* d_ws, size_t ws_size,
                              hipStream_t stream) {
  (void)in_sizes; (void)n_in; (void)out_size; (void)ws_size;
  const float* masks = (const float*)d_in[0];
  float* out  = (float*)d_out;
  float* skel = out;          // first half: edt2 scratch -> skeleton
  float* f2   = out + HALF;   // second half: f2 scratch -> boundary
  unsigned int* mx = (unsigned int*)d_ws;

  k_init   <<<1, 32, 0, stream>>>(mx);
  k_rowpass<<<(Bn * Hn) / 8, 256, 0, stream>>>(masks, f2);
  k_colpass<<<dim3(Wn / 64, Hn / 128, Bn), 512, Hn * 64 * sizeof(float), stream>>>(f2, skel, mx);
  k_norm   <<<HALF / 256, 256, 0, stream>>>(masks, mx, skel, f2);
}